// ConformalMetric_5360119185650
// MI455X (gfx1250) — compile-verified
//
#include <hip/hip_runtime.h>

// Problem dims (fixed by the reference): (B,H,W,D) = (4,128,128,128), fp32.
#define BB 4
#define HH 128
#define WW 128
#define DD 128

// Tile: 2 (h) x 4 (w) x full 128 (d). Halo only in h/w (z row is complete,
// so the edge-clamp in z is resolved inside the tile).
#define TH 2
#define TW 4
#define HALO_H (TH + 2)                 // 4
#define HALO_W (TW + 2)                 // 6
#define TILE_ROWS (HALO_H * HALO_W)     // 24 rows of 128 floats
#define TILE_ELEMS (TILE_ROWS * DD)     // 3072 floats = 12 KB
#define NTHREADS 256

// 128-bit chunk type expected by the async-copy builtin.
typedef int v4i __attribute__((ext_vector_type(4)));

// ---- gfx1250 async global->LDS path (guarded; fallback = direct loads) ----
#if defined(__gfx1250__) && __has_builtin(__builtin_amdgcn_global_load_async_to_lds_b128)
  #define HAVE_ASYNC 1
  #define ASYNC_B128(gp, lp)                                                   \
    __builtin_amdgcn_global_load_async_to_lds_b128(                            \
        (__attribute__((address_space(1))) v4i*)(gp),                          \
        (__attribute__((address_space(3))) v4i*)(lp), 0, 0)
#else
  #define HAVE_ASYNC 0
#endif

#if defined(__gfx1250__) && __has_builtin(__builtin_amdgcn_s_wait_asynccnt)
  #define WAIT_ASYNC() __builtin_amdgcn_s_wait_asynccnt(0)
#elif HAVE_ASYNC
  #define WAIT_ASYNC() asm volatile("s_wait_asynccnt 0x0" ::: "memory")
#else
  #define WAIT_ASYNC()
#endif

__global__ __launch_bounds__(NTHREADS) void ConformalMetric_stencil_kernel(
    const float* __restrict__ pos, const float* __restrict__ neg,
    const float* __restrict__ lam_p, float* __restrict__ out) {
  // Phi tile with h/w halo; with the async path we also stage raw phi_negative.
  __shared__ __attribute__((aligned(16))) float sP[TILE_ELEMS];
#if HAVE_ASYNC
  __shared__ __attribute__((aligned(16))) float sN[TILE_ELEMS];
#endif

  const int tid = threadIdx.x;
  const int w0 = blockIdx.x * TW;
  const int h0 = blockIdx.y * TH;
  const int b = blockIdx.z;
  const float lam = lam_p[0];  // scalar load (SMEM), uniform

  // ---- Stage haloed tile: 768 float4 chunks per array, coalesced in d ----
  const int nvec = TILE_ELEMS / 4;  // 768
  for (int t = tid; t < nvec; t += NTHREADS) {
    const int d4 = (t & 31) << 2;       // d offset in floats (16B chunks)
    const int row = t >> 5;             // 0..23
    const int lh = row / HALO_W;
    const int lw = row % HALO_W;
    // Edge-clamped global coordinates implement the 'edge' padding.
    const int gh = min(max(h0 + lh - 1, 0), HH - 1);
    const int gw = min(max(w0 + lw - 1, 0), WW - 1);
    const size_t g = ((((size_t)b * HH + gh) * WW + gw) * DD) + d4;
    const int l = row * DD + d4;
#if HAVE_ASYNC
    ASYNC_B128(pos + g, &sP[l]);        // ASYNCcnt-tracked, no VGPR staging
    ASYNC_B128(neg + g, &sN[l]);
#else
    const float4 p4 = *(const float4*)(pos + g);
    const float4 n4 = *(const float4*)(neg + g);
    float4 c4;
    c4.x = fmaf(-lam, n4.x, p4.x);
    c4.y = fmaf(-lam, n4.y, p4.y);
    c4.z = fmaf(-lam, n4.z, p4.z);
    c4.w = fmaf(-lam, n4.w, p4.w);
    *(float4*)&sP[l] = c4;
#endif
  }

#if HAVE_ASYNC
  WAIT_ASYNC();  // own wave's async copies landed in LDS
  // In-place elementwise combine Phi = pos - lam*neg over *this thread's own*
  // staged chunks -> no barrier needed before this pass.
  for (int t = tid; t < nvec; t += NTHREADS) {
    const int l = (t >> 5) * DD + ((t & 31) << 2);
    const float4 p4 = *(const float4*)&sP[l];
    const float4 n4 = *(const float4*)&sN[l];
    float4 c4;
    c4.x = fmaf(-lam, n4.x, p4.x);
    c4.y = fmaf(-lam, n4.y, p4.y);
    c4.z = fmaf(-lam, n4.z, p4.z);
    c4.w = fmaf(-lam, n4.w, p4.w);
    *(float4*)&sP[l] = c4;
  }
#endif

  __syncthreads();  // all waves' Phi tile visible (s_wait_dscnt + barrier)

  // ---- 7-point stencil from LDS: 1024 interior elems, 4 per thread ----
  for (int e = tid; e < TH * TW * DD; e += NTHREADS) {
    const int d = e & (DD - 1);
    const int row = e >> 7;             // 0..7
    const int lw = row & (TW - 1);      // 0..3
    const int lh = row >> 2;            // 0..1
    const int ci = (lh + 1) * (HALO_W * DD) + (lw + 1) * DD + d;

    const float c = sP[ci];
    const float xp = sP[ci + HALO_W * DD];          // h+1
    const float xm = sP[ci - HALO_W * DD];          // h-1
    const float yp = sP[ci + DD];                   // w+1
    const float ym = sP[ci - DD];                   // w-1
    const float zp = sP[ci + ((d == DD - 1) ? 0 : 1)];  // z edge-clamped
    const float zm = sP[ci - ((d == 0) ? 0 : 1)];

    const float gx = (xp - xm) * 0.5f;  // inv(2h), h = 1
    const float gy = (yp - ym) * 0.5f;
    const float gz = (zp - zm) * 0.5f;
    const float gsq = gx * gx + gy * gy + gz * gz;
    const float lap = (xp + xm + yp + ym + zp + zm) - 6.0f * c;  // /h^2, h=1
    const float R = -2.0f * __expf(-2.0f * c) * fmaf(2.0f, gsq, lap);

    const size_t g = ((((size_t)b * HH + (h0 + lh)) * WW + (w0 + lw)) * DD) + d;
    __builtin_nontemporal_store(R, out + g);  // streaming write, NT hint
  }
}

extern "C" void kernel_launch(void* const* d_in, const int* in_sizes, int n_in,
                              void* d_out, int out_size, void* d_ws, size_t ws_size,
                              hipStream_t stream) {
  const float* pos = (const float*)d_in[0];   // phi_positive
  const float* neg = (const float*)d_in[1];   // phi_negative
  const float* lam = (const float*)d_in[2];   // lambda_repulsion (1 elem)
  float* out = (float*)d_out;

  dim3 grid(WW / TW, HH / TH, BB);  // 32 x 64 x 4 = 8192 blocks
  ConformalMetric_stencil_kernel<<<grid, NTHREADS, 0, stream>>>(pos, neg, lam, out);
}